// DecoderBiRNN_66932770341216
// MI455X (gfx1250) — compile-verified
//
#include <hip/hip_runtime.h>
#include <cmath>

#define H     1024
#define LSEQ  2048
#define VSZ   50257
#define CH    4096   // 4*H concat length

typedef __attribute__((ext_vector_type(2))) float v2f;
typedef __attribute__((ext_vector_type(8))) float v8f;

// workspace layout (float offsets)
#define WS_H   0        // 2048   h (2,1024)
#define WS_E   2048     // 4096   energy (2,2048)
#define WS_A   6144     // 4096   attn (2,2048)
#define WS_C   10240    // 4096   concat (4096)
#define WS_P   14336    // 32768  attn_out partials (2,16,1024)
#define WS_R   47104    // 2      {max, log(sumexp)}

// ---------------------------------------------------------------------------
// Kernel 1: RNN cell  h = tanh(W_ih@x + b_ih + W_hh@h0 + b_hh)
// one wave per output row (2048 rows), float4 coalesced loads, shfl reduce
__global__ __launch_bounds__(256) void k_rnn(const int* __restrict__ tok,
    const float* __restrict__ hidden, const float* __restrict__ emb,
    const float* __restrict__ W_ih, const float* __restrict__ W_hh,
    const float* __restrict__ b_ih, const float* __restrict__ b_hh,
    float* __restrict__ hws, float* __restrict__ hout) {
  int wave = threadIdx.x >> 5, lane = threadIdx.x & 31;
  int r = blockIdx.x * 8 + wave;           // 0..2047  (r = d*1024 + i)
  int d = r >> 10;
  const float4* x4  = (const float4*)(emb + (size_t)tok[0] * H);
  const float4* h04 = (const float4*)(hidden + d * H);
  const float4* wi4 = (const float4*)(W_ih + (size_t)r * H);
  const float4* wh4 = (const float4*)(W_hh + (size_t)r * H);
  float acc = 0.f;
  #pragma unroll
  for (int o = lane; o < H / 4; o += 32) {
    float4 a = wi4[o], b = x4[o], c = wh4[o], e = h04[o];
    acc += a.x * b.x + a.y * b.y + a.z * b.z + a.w * b.w;
    acc += c.x * e.x + c.y * e.y + c.z * e.z + c.w * e.w;
  }
  for (int m = 16; m; m >>= 1) acc += __shfl_xor(acc, m, 32);
  if (lane == 0) {
    float v = tanhf(acc + b_ih[r] + b_hh[r]);
    hws[r]  = v;    // workspace copy
    hout[r] = v;    // second output: h[:,None,:] -> d_out + VSZ
  }
}

// ---------------------------------------------------------------------------
// Kernel 2: energy[d][l] = dot(h[d], enc[l]); one wave per (d,l)
__global__ __launch_bounds__(256) void k_energy(const float* __restrict__ enc,
    const float* __restrict__ hws, float* __restrict__ energy) {
  int wave = threadIdx.x >> 5, lane = threadIdx.x & 31;
  int r = blockIdx.x * 8 + wave;           // 0..4095
  int d = r >> 11, l = r & (LSEQ - 1);
  const float4* e4 = (const float4*)(enc + (size_t)l * H);
  const float4* h4 = (const float4*)(hws + d * H);
  float acc = 0.f;
  #pragma unroll
  for (int o = lane; o < H / 4; o += 32) {
    float4 a = e4[o], b = h4[o];
    acc += a.x * b.x + a.y * b.y + a.z * b.z + a.w * b.w;
  }
  for (int m = 16; m; m >>= 1) acc += __shfl_xor(acc, m, 32);
  if (lane == 0) energy[(size_t)d * LSEQ + l] = acc;
}

// ---------------------------------------------------------------------------
// Kernel 3: softmax over l for both directions (single block)
__global__ __launch_bounds__(256) void k_softmax(const float* __restrict__ energy,
                                                 float* __restrict__ attn) {
  __shared__ float red[256];
  for (int d = 0; d < 2; ++d) {
    const float* e = energy + (size_t)d * LSEQ;
    float*       a = attn   + (size_t)d * LSEQ;
    float mx = -1e30f;
    for (int l = threadIdx.x; l < LSEQ; l += 256) mx = fmaxf(mx, e[l]);
    red[threadIdx.x] = mx; __syncthreads();
    for (int s = 128; s; s >>= 1) {
      if (threadIdx.x < s) red[threadIdx.x] = fmaxf(red[threadIdx.x], red[threadIdx.x + s]);
      __syncthreads();
    }
    mx = red[0]; __syncthreads();
    float sm = 0.f;
    for (int l = threadIdx.x; l < LSEQ; l += 256) sm += expf(e[l] - mx);
    red[threadIdx.x] = sm; __syncthreads();
    for (int s = 128; s; s >>= 1) {
      if (threadIdx.x < s) red[threadIdx.x] += red[threadIdx.x + s];
      __syncthreads();
    }
    float inv = 1.f / red[0]; __syncthreads();
    for (int l = threadIdx.x; l < LSEQ; l += 256) a[l] = expf(e[l] - mx) * inv;
  }
}

// ---------------------------------------------------------------------------
// Kernel 4a: attn_out partials over 16 l-chunks (deterministic, no atomics)
__global__ __launch_bounds__(256) void k_attnout_part(const float* __restrict__ enc,
    const float* __restrict__ attn, float* __restrict__ part) {
  int i  = blockIdx.x * 256 + threadIdx.x;  // 0..1023
  int lc = blockIdx.y;                      // 0..15
  float a0 = 0.f, a1 = 0.f;
  int l0 = lc * (LSEQ / 16);
  #pragma unroll 4
  for (int l = l0; l < l0 + LSEQ / 16; ++l) {
    float e = enc[(size_t)l * H + i];
    a0 += attn[l] * e;
    a1 += attn[LSEQ + l] * e;
  }
  part[(size_t)lc * H + i]        = a0;
  part[(size_t)(16 + lc) * H + i] = a1;
}

// Kernel 4b: reduce partials, build concat = [ao0, h0, ao1, h1]
__global__ __launch_bounds__(256) void k_concat(const float* __restrict__ part,
    const float* __restrict__ hws, float* __restrict__ concat) {
  int i = blockIdx.x * 256 + threadIdx.x;   // 0..1023
  float a0 = 0.f, a1 = 0.f;
  #pragma unroll
  for (int c = 0; c < 16; ++c) {
    a0 += part[(size_t)c * H + i];
    a1 += part[(size_t)(16 + c) * H + i];
  }
  concat[i]         = a0;
  concat[H + i]     = hws[i];
  concat[2 * H + i] = a1;
  concat[3 * H + i] = hws[H + i];
}

// ---------------------------------------------------------------------------
// Kernel 5: logits = W_out(50257x4096) @ concat + b_out  via fp32 WMMA.
// A (16x4) = concat chunk broadcast over all 16 M rows:
//   lanes 0-15 hold K=0,1 ; lanes 16-31 hold K=2,3 (ISA 32-bit A layout).
// B (4x16)  = W_out tile, N = lane%16 output row, same K split per half-wave.
// D[m][n] identical over m -> lane L's c[0] = logits partial for row v0+L%16.
__global__ __launch_bounds__(256) void k_logits_wmma(const float* __restrict__ Wout,
    const float* __restrict__ b_out, const float* __restrict__ concat,
    float* __restrict__ logits) {
  __shared__ float sx[CH];
  for (int i = threadIdx.x; i < CH; i += 256) sx[i] = concat[i];
  __syncthreads();

  int wave = threadIdx.x >> 5, lane = threadIdx.x & 31;
  int half = lane >> 4;                 // 0: K=0,1  1: K=2,3
  int n    = lane & 15;
  long v0  = (long)blockIdx.x * 128 + (long)wave * 16;
  if (v0 >= VSZ) return;                // wave-uniform exit; EXEC stays all-1s
  long row = v0 + n;
  if (row >= VSZ) row = VSZ - 1;        // clamp tail row (no divergence)
  const float* wrow = Wout + row * (size_t)CH + 2 * half;
  const float* sxh  = sx + 2 * half;

  v8f c0 = {}, c1 = {}, c2 = {}, c3 = {};
  #pragma unroll 2
  for (int o = 0; o < CH; o += 16) {
    v2f a0 = *(const v2f*)(sxh + o);
    v2f a1 = *(const v2f*)(sxh + o + 4);
    v2f a2 = *(const v2f*)(sxh + o + 8);
    v2f a3 = *(const v2f*)(sxh + o + 12);
    v2f b0 = __builtin_nontemporal_load((const v2f*)(wrow + o));
    v2f b1 = __builtin_nontemporal_load((const v2f*)(wrow + o + 4));
    v2f b2 = __builtin_nontemporal_load((const v2f*)(wrow + o + 8));
    v2f b3 = __builtin_nontemporal_load((const v2f*)(wrow + o + 12));
    c0 = __builtin_amdgcn_wmma_f32_16x16x4_f32(false, a0, false, b0, (short)0, c0, false, false);
    c1 = __builtin_amdgcn_wmma_f32_16x16x4_f32(false, a1, false, b1, (short)0, c1, false, false);
    c2 = __builtin_amdgcn_wmma_f32_16x16x4_f32(false, a2, false, b2, (short)0, c2, false, false);
    c3 = __builtin_amdgcn_wmma_f32_16x16x4_f32(false, a3, false, b3, (short)0, c3, false, false);
  }
  v8f cs = (c0 + c1) + (c2 + c3);
  if (lane < 16 && v0 + lane < VSZ)
    logits[v0 + lane] = cs[0] + b_out[v0 + lane];
}

// ---------------------------------------------------------------------------
// Kernel 6: log-softmax reduction (max, log-sum-exp) over VSZ logits
__global__ __launch_bounds__(1024) void k_lsm_reduce(const float* __restrict__ logits,
                                                     float* __restrict__ red) {
  __shared__ float s[1024];
  float mx = -1e30f;
  for (int v = threadIdx.x; v < VSZ; v += 1024) mx = fmaxf(mx, logits[v]);
  s[threadIdx.x] = mx; __syncthreads();
  for (int st = 512; st; st >>= 1) {
    if (threadIdx.x < st) s[threadIdx.x] = fmaxf(s[threadIdx.x], s[threadIdx.x + st]);
    __syncthreads();
  }
  mx = s[0]; __syncthreads();
  float sm = 0.f;
  for (int v = threadIdx.x; v < VSZ; v += 1024) sm += expf(logits[v] - mx);
  s[threadIdx.x] = sm; __syncthreads();
  for (int st = 512; st; st >>= 1) {
    if (threadIdx.x < st) s[threadIdx.x] += s[threadIdx.x + st];
    __syncthreads();
  }
  if (threadIdx.x == 0) { red[0] = mx; red[1] = logf(s[0]); }
}

// Kernel 7: in-place normalize: out[v] = logits[v] - max - lse
__global__ __launch_bounds__(256) void k_lsm_apply(float* __restrict__ out,
                                                   const float* __restrict__ red) {
  int v = blockIdx.x * 256 + threadIdx.x;
  if (v < VSZ) out[v] = out[v] - red[0] - red[1];
}

// ---------------------------------------------------------------------------
extern "C" void kernel_launch(void* const* d_in, const int* in_sizes, int n_in,
                              void* d_out, int out_size, void* d_ws, size_t ws_size,
                              hipStream_t stream) {
  const int*   tok    = (const int*)  d_in[0];
  const float* hidden = (const float*)d_in[1];
  const float* enc    = (const float*)d_in[2];
  const float* emb    = (const float*)d_in[3];
  const float* W_ih   = (const float*)d_in[4];
  const float* W_hh   = (const float*)d_in[5];
  const float* b_ih   = (const float*)d_in[6];
  const float* b_hh   = (const float*)d_in[7];
  const float* W_out  = (const float*)d_in[8];
  const float* b_out  = (const float*)d_in[9];

  float* out = (float*)d_out;            // [0,VSZ): log-probs; [VSZ,VSZ+2048): h
  float* ws  = (float*)d_ws;

  float* hws    = ws + WS_H;
  float* energy = ws + WS_E;
  float* attn   = ws + WS_A;
  float* concat = ws + WS_C;
  float* part   = ws + WS_P;
  float* red    = ws + WS_R;

  // 1) RNN cell: 2048 rows, one wave each
  k_rnn<<<2048 / 8, 256, 0, stream>>>(tok, hidden, emb, W_ih, W_hh, b_ih, b_hh,
                                      hws, out + VSZ);
  // 2) attention energies: 2*2048 rows
  k_energy<<<4096 / 8, 256, 0, stream>>>(enc, hws, energy);
  // 3) softmax over l for both directions
  k_softmax<<<1, 256, 0, stream>>>(energy, attn);
  // 4) attn_out partials + concat assembly
  k_attnout_part<<<dim3(H / 256, 16), 256, 0, stream>>>(enc, attn, part);
  k_concat<<<H / 256, 256, 0, stream>>>(part, hws, concat);
  // 5) big GEMV via fp32 WMMA: 16 rows/wave, 128 rows/block
  k_logits_wmma<<<(VSZ + 127) / 128, 256, 0, stream>>>(W_out, b_out, concat, out);
  // 6/7) log-softmax
  k_lsm_reduce<<<1, 1024, 0, stream>>>(out, red);
  k_lsm_apply<<<(VSZ + 255) / 256, 256, 0, stream>>>(out, red);
}